// BuildVolume_70437463654734
// MI455X (gfx1250) — compile-verified
//
#include <hip/hip_runtime.h>

typedef float v2f __attribute__((ext_vector_type(2)));
typedef float v8f __attribute__((ext_vector_type(8)));

#define Hdim   128
#define Wdim   128
#define Ndim   4
#define Cdim   8
#define Vdim   81
#define Mdim   9
#define OUTdim 64
#define Kdim   (Cdim * Vdim)          // 648
#define PIX    (Hdim * Wdim * Ndim)   // 65536
#define TILE_P 128

__global__ __launch_bounds__(128)
void build_volume_wmma(const float* __restrict__ deltmap,   // [H,W,N]
                       const float* __restrict__ imageMxM,  // [H,W,C,M,M]
                       const float* __restrict__ x_g,       // [H,W,V]
                       const float* __restrict__ conv_w,    // [OUT, K], k = c*V + v
                       const float* __restrict__ conv_b,    // [OUT]
                       float* __restrict__ out)             // [OUT, H, W, N]
{
    __shared__ float lds[Cdim][TILE_P + 4];   // per-view gated cost slice

    const int t    = threadIdx.x;
    const int wave = t >> 5;          // 0..3 -> output row block
    const int lane = t & 31;
    const int pb   = blockIdx.x * TILE_P;

    // 8 pixel-blocks of 16, each a 16x16 f32 accumulator tile (8 VGPRs/lane)
    v8f acc[8];
    #pragma unroll
    for (int i = 0; i < 8; ++i) acc[i] = (v8f)0.0f;

    // ---- per-thread pixel (sampling phase) ----
    const int p = pb + t;
    const int n = p & (Ndim - 1);
    const int w = (p >> 2) & (Wdim - 1);
    const int h = p >> 9;
    (void)n;
    const float delt   = deltmap[p];
    const float base_x = -1.0f + 2.0f * (float)w / (float)(Wdim - 1);
    const float base_y = -1.0f + 2.0f * (float)h / (float)(Hdim - 1);
    const float* gate_row = x_g + (size_t)(h * Wdim + w) * Vdim;

    const int m16   = lane & 15;
    const int khalf = lane >> 4;      // 0: K rows {0,1}; 1: K rows {2,3}
    const int c0    = 2 * khalf;

    for (int v = 0; v < Vdim; ++v) {
        // ================= Phase A: gated bilinear sample -> LDS =================
        const int   uu = v % Mdim;
        const int   vv = v / Mdim;
        const float sx = 2.0f * (float)(Mdim / 2 - uu) / (float)Wdim;
        const float sy = 2.0f * (float)(Mdim / 2 - vv) / (float)Hdim;
        const float gx = base_x + sx * delt;
        const float gy = base_y + sy * delt;
        const float ixf = ((gx + 1.0f) * (float)Wdim - 1.0f) * 0.5f;  // align_corners=False
        const float iyf = ((gy + 1.0f) * (float)Hdim - 1.0f) * 0.5f;
        const float fx0 = floorf(ixf);
        const float fy0 = floorf(iyf);
        const int x0 = (int)fx0, x1 = x0 + 1;
        const int y0 = (int)fy0, y1 = y0 + 1;
        const float wx1 = ixf - fx0, wx0 = 1.0f - wx1;
        const float wy1 = iyf - fy0, wy0 = 1.0f - wy1;

        float ch[Cdim];
        #pragma unroll
        for (int c = 0; c < Cdim; ++c) ch[c] = 0.0f;

        #pragma unroll
        for (int corner = 0; corner < 4; ++corner) {
            const int xi = (corner & 1) ? x1 : x0;
            const int yi = (corner & 2) ? y1 : y0;
            float wgt = ((corner & 1) ? wx1 : wx0) * ((corner & 2) ? wy1 : wy0);
            const bool valid = (xi >= 0) && (xi < Wdim) && (yi >= 0) && (yi < Hdim);
            wgt = valid ? wgt : 0.0f;                      // padding_mode='zeros'
            const int xc = min(max(xi, 0), Wdim - 1);
            const int yc = min(max(yi, 0), Hdim - 1);
            // imageMxM[yc, xc, c, vv, uu], channels strided by V; L2-resident
            const float* src = imageMxM + (size_t)(yc * Wdim + xc) * (Cdim * Vdim) + v;
            #pragma unroll
            for (int c = 0; c < Cdim; ++c)
                ch[c] += wgt * src[c * Vdim];
        }
        const float gate = gate_row[v];
        #pragma unroll
        for (int c = 0; c < Cdim; ++c)
            lds[c][t] = gate * ch[c];

        __syncthreads();

        // ================= Phase B: WMMA over the 8 k-rows of this view ==========
        // A fragment (16x4 f32 ISA layout): lane holds A[m16][c0+j], j=0,1
        // weight element for (o, k' = v*8 + c) is conv_w[o*K + c*V + v]
        const float* wrow = conv_w + (size_t)(wave * 16 + m16) * Kdim + v;
        v2f a0, a1;
        a0.x = wrow[(c0    ) * Vdim];
        a0.y = wrow[(c0 + 1) * Vdim];
        a1.x = wrow[(c0 + 4) * Vdim];
        a1.y = wrow[(c0 + 5) * Vdim];

        #pragma unroll
        for (int blk = 0; blk < 8; ++blk) {
            const int col = blk * 16 + m16;
            v2f b0, b1;                       // B 4x16: lane holds B[c0+j][m16]
            b0.x = lds[c0    ][col];
            b0.y = lds[c0 + 1][col];
            b1.x = lds[c0 + 4][col];
            b1.y = lds[c0 + 5][col];
            acc[blk] = __builtin_amdgcn_wmma_f32_16x16x4_f32(
                false, a0, false, b0, (short)0, acc[blk], false, false);
            acc[blk] = __builtin_amdgcn_wmma_f32_16x16x4_f32(
                false, a1, false, b1, (short)0, acc[blk], false, false);
        }
        __syncthreads();
    }

    // ================= Epilogue: bias + store ===================================
    // C/D layout: VGPR r -> M = r + 8*khalf, lane column = m16
    #pragma unroll
    for (int r = 0; r < 8; ++r) {
        const int o = wave * 16 + 8 * khalf + r;
        const float bias = conv_b[o];
        float* orow = out + (size_t)o * PIX + pb;
        #pragma unroll
        for (int blk = 0; blk < 8; ++blk)
            orow[blk * 16 + m16] = acc[blk][r] + bias;
    }
}

extern "C" void kernel_launch(void* const* d_in, const int* in_sizes, int n_in,
                              void* d_out, int out_size, void* d_ws, size_t ws_size,
                              hipStream_t stream) {
    (void)in_sizes; (void)n_in; (void)d_ws; (void)ws_size; (void)out_size;
    const float* deltmap  = (const float*)d_in[0];
    const float* imageMxM = (const float*)d_in[1];
    const float* x_g      = (const float*)d_in[2];
    const float* conv_w   = (const float*)d_in[3];
    const float* conv_b   = (const float*)d_in[4];
    float* out = (float*)d_out;

    dim3 grid(PIX / TILE_P);   // 512 workgroups
    dim3 block(TILE_P);        // 128 threads = 4 wave32
    build_volume_wmma<<<grid, block, 0, stream>>>(deltmap, imageMxM, x_g,
                                                  conv_w, conv_b, out);
}